// KGAN_71425306133078
// MI455X (gfx1250) — compile-verified
//
#include <hip/hip_runtime.h>
#include <hip/hip_bf16.h>

typedef __attribute__((ext_vector_type(2))) float v2f;
typedef __attribute__((ext_vector_type(4))) float f4;
typedef __attribute__((ext_vector_type(8))) float v8f;

#define N_ENT 100000
#define KNB   32
#define ED    128
#define BSZ   16384
#define PAD   132   // padded LDS row stride (floats): (4*row + k) % 64 conflict-free

// ---------------------------------------------------------------------------
// Kernel A: per batch element -> gather, max-norm, WMMA GEMM tr = [t;h]@Wr^T,
// tanh attention, softmax, aggregate; write h+Nh and h*Nh to workspace.
// Block = 128 threads (4 wave32), grid = B.
// ---------------------------------------------------------------------------
__global__ __launch_bounds__(128) void kgan_attn(
    const int* __restrict__ idx, const int* __restrict__ adj_ent,
    const int* __restrict__ adj_rel, const float* __restrict__ ent_table,
    const float* __restrict__ rel_table, const float* __restrict__ Wr_w,
    const float* __restrict__ Wr_b, float* __restrict__ hpN,
    float* __restrict__ hmN)
{
    extern __shared__ float smem[];
    float* Wr_lds = smem;                    // 128*PAD
    float* t_lds  = Wr_lds + 128 * PAD;      // 48*PAD  (rows 0..31 = t, 32 = h, 33..47 = 0)
    float* tr_lds = t_lds  + 48  * PAD;      // 48*PAD
    float* r_lds  = tr_lds + 48  * PAD;      // 32*PAD
    float* bias   = r_lds  + 32  * PAD;      // 128
    float* sp     = bias + 128;              // 32*4 partial scores
    float* att    = sp + 128;                // 32
    int*   eids   = (int*)(att + 32);        // 32
    int*   rids   = eids + 32;               // 32

    const int tid  = threadIdx.x;
    const int wave = tid >> 5;
    const int lane = tid & 31;
    const int b    = blockIdx.x;

    int id = idx[b];
    id = id < 0 ? 0 : (id > N_ENT - 1 ? N_ENT - 1 : id);

    if (tid < KNB) {
        eids[tid] = adj_ent[(size_t)id * KNB + tid];
        rids[tid] = adj_rel[(size_t)id * KNB + tid];
    }
    bias[tid] = Wr_b[tid];
    // zero pad rows 33..47 of the A matrix
    for (int i = tid; i < 15 * ED; i += 128) {
        int r = 33 + i / ED, c = i % ED;
        t_lds[r * PAD + c] = 0.0f;
    }
    // stage Wr (row-major [j][d]) into LDS, padded
    for (int i = tid; i < 128 * 32; i += 128) {
        int row = i >> 5, c4 = i & 31;
        *(f4*)(&Wr_lds[row * PAD + 4 * c4]) = ((const f4*)Wr_w)[row * 32 + c4];
    }
    __syncthreads();

    // max-norm renormalized gathers: jobs 0..31 = t rows, 32..63 = r rows, 64 = h
    for (int job = wave; job < 65; job += 4) {
        const float* src;
        float* dst;
        if (job < 32)      { src = ent_table + (size_t)eids[job] * ED;      dst = t_lds + job * PAD; }
        else if (job < 64) { src = rel_table + (size_t)rids[job - 32] * ED; dst = r_lds + (job - 32) * PAD; }
        else               { src = ent_table + (size_t)id * ED;             dst = t_lds + 32 * PAD; }
        f4 v = ((const f4*)src)[lane];
        float ss = v.x * v.x + v.y * v.y + v.z * v.z + v.w * v.w;
        #pragma unroll
        for (int m = 16; m >= 1; m >>= 1) ss += __shfl_xor(ss, m, 32);
        float n = sqrtf(ss);
        float s = fminf(1.0f, 1.0f / fmaxf(n, 1e-12f));
        v.x *= s; v.y *= s; v.z *= s; v.w *= s;
        *(f4*)(&dst[lane * 4]) = v;
    }
    __syncthreads();

    // ---- fp32 WMMA GEMM: tr[0..47][0..127] = [t;h;0] @ Wr^T --------------
    const int koff = (lane >> 4) << 1;   // lanes 0-15 -> K {0,1}, 16-31 -> K {2,3}
    const int mrow = lane & 15;
    const int nt0  = wave * 2;           // each wave owns 2 N-tiles x 3 M-tiles
    v8f vzero = {};
    v8f acc[2][3];
    #pragma unroll
    for (int i = 0; i < 2; ++i)
        #pragma unroll
        for (int mt = 0; mt < 3; ++mt) acc[i][mt] = vzero;

    for (int kk = 0; kk < 32; ++kk) {
        const int k = kk * 4 + koff;
        v2f a[3], bf[2];
        #pragma unroll
        for (int mt = 0; mt < 3; ++mt) {
            const float* p = &t_lds[(mt * 16 + mrow) * PAD + k];
            a[mt].x = p[0]; a[mt].y = p[1];
        }
        #pragma unroll
        for (int i = 0; i < 2; ++i) {
            const float* p = &Wr_lds[((nt0 + i) * 16 + mrow) * PAD + k];
            bf[i].x = p[0]; bf[i].y = p[1];
        }
        #pragma unroll
        for (int i = 0; i < 2; ++i)
            #pragma unroll
            for (int mt = 0; mt < 3; ++mt)
                acc[i][mt] = __builtin_amdgcn_wmma_f32_16x16x4_f32(
                    false, a[mt], false, bf[i], (short)0, acc[i][mt], false, false);
    }
    // write D fragments (+bias) to tr_lds
    #pragma unroll
    for (int i = 0; i < 2; ++i)
        #pragma unroll
        for (int mt = 0; mt < 3; ++mt)
            #pragma unroll
            for (int v = 0; v < 8; ++v) {
                int row = mt * 16 + v + ((lane >> 4) << 3);
                int col = (nt0 + i) * 16 + (lane & 15);
                tr_lds[row * PAD + col] = acc[i][mt][v] + bias[col];
            }
    __syncthreads();

    // ---- attention scores: score[k] = sum_j tanh(hr[j]+r[k][j]) * tr[k][j]
    {
        int k = tid >> 2, q = tid & 3;
        const float* hr = &tr_lds[32 * PAD];
        float s = 0.0f;
        for (int j = q * 32; j < q * 32 + 32; ++j)
            s += tanhf(hr[j] + r_lds[k * PAD + j]) * tr_lds[k * PAD + j];
        sp[k * 4 + q] = s;
    }
    __syncthreads();
    if (wave == 0) {
        float sc = sp[lane * 4] + sp[lane * 4 + 1] + sp[lane * 4 + 2] + sp[lane * 4 + 3];
        float mx = sc;
        #pragma unroll
        for (int m = 16; m >= 1; m >>= 1) mx = fmaxf(mx, __shfl_xor(mx, m, 32));
        float e = expf(sc - mx);
        float sum = e;
        #pragma unroll
        for (int m = 16; m >= 1; m >>= 1) sum += __shfl_xor(sum, m, 32);
        att[lane] = e / sum;
    }
    __syncthreads();

    // ---- aggregate: Nh[d] = sum_k att[k]*t[k][d]; emit h+Nh, h*Nh --------
    {
        int d = tid;
        float nh = 0.0f;
        #pragma unroll
        for (int k2 = 0; k2 < 32; ++k2) nh += att[k2] * t_lds[k2 * PAD + d];
        float h = t_lds[32 * PAD + d];
        hpN[(size_t)b * ED + d] = h + nh;
        hmN[(size_t)b * ED + d] = h * nh;
    }
}

// ---------------------------------------------------------------------------
// Kernel B: out = leaky(hpN@W1^T+b1) + leaky(hmN@W2^T+b2)
// Block = 128 threads (4 waves), 64 output rows per block; two WMMA passes.
// ---------------------------------------------------------------------------
__global__ __launch_bounds__(128) void kgan_out(
    const float* __restrict__ hpN, const float* __restrict__ hmN,
    const float* __restrict__ W1_w, const float* __restrict__ W1_b,
    const float* __restrict__ W2_w, const float* __restrict__ W2_b,
    float* __restrict__ out)
{
    extern __shared__ float smem[];
    float* W_lds = smem;                // 128*PAD
    float* A_lds = W_lds + 128 * PAD;   // 64*PAD
    float* bias  = A_lds + 64 * PAD;    // 128

    const int tid  = threadIdx.x;
    const int wave = tid >> 5;
    const int lane = tid & 31;
    const int m0   = blockIdx.x * 64;

    float res[8][8];
    v8f vzero = {};

    for (int pass = 0; pass < 2; ++pass) {
        const float* W  = pass ? W2_w : W1_w;
        const float* bb = pass ? W2_b : W1_b;
        const float* A  = pass ? hmN : hpN;
        if (pass) __syncthreads();   // pass-0 readers done before overwrite
        for (int i = tid; i < 128 * 32; i += 128) {
            int row = i >> 5, c4 = i & 31;
            *(f4*)(&W_lds[row * PAD + 4 * c4]) = ((const f4*)W)[row * 32 + c4];
        }
        for (int i = tid; i < 64 * 32; i += 128) {
            int row = i >> 5, c4 = i & 31;
            *(f4*)(&A_lds[row * PAD + 4 * c4]) =
                ((const f4*)(A + (size_t)m0 * ED))[row * 32 + c4];
        }
        bias[tid] = bb[tid];
        __syncthreads();

        const int koff = (lane >> 4) << 1;
        const int mrow = lane & 15;
        v8f acc[8];
        #pragma unroll
        for (int i = 0; i < 8; ++i) acc[i] = vzero;

        for (int kk = 0; kk < 32; ++kk) {
            const int k = kk * 4 + koff;
            v2f a;
            const float* pa = &A_lds[(wave * 16 + mrow) * PAD + k];
            a.x = pa[0]; a.y = pa[1];
            #pragma unroll
            for (int nt = 0; nt < 8; ++nt) {
                v2f bf;
                const float* pb = &W_lds[(nt * 16 + mrow) * PAD + k];
                bf.x = pb[0]; bf.y = pb[1];
                acc[nt] = __builtin_amdgcn_wmma_f32_16x16x4_f32(
                    false, a, false, bf, (short)0, acc[nt], false, false);
            }
        }
        #pragma unroll
        for (int nt = 0; nt < 8; ++nt)
            #pragma unroll
            for (int v = 0; v < 8; ++v) {
                int col = nt * 16 + (lane & 15);
                float x = acc[nt][v] + bias[col];
                x = (x >= 0.0f) ? x : 0.2f * x;
                if (pass == 0) res[nt][v] = x; else res[nt][v] += x;
            }
    }

    #pragma unroll
    for (int nt = 0; nt < 8; ++nt)
        #pragma unroll
        for (int v = 0; v < 8; ++v) {
            int row = m0 + wave * 16 + v + ((lane >> 4) << 3);
            int col = nt * 16 + (lane & 15);
            out[(size_t)row * ED + col] = res[nt][v];
        }
}

extern "C" void kernel_launch(void* const* d_in, const int* in_sizes, int n_in,
                              void* d_out, int out_size, void* d_ws, size_t ws_size,
                              hipStream_t stream) {
    (void)in_sizes; (void)n_in; (void)out_size; (void)ws_size;
    const int*   idx       = (const int*)d_in[0];
    const int*   adj_ent   = (const int*)d_in[1];
    const int*   adj_rel   = (const int*)d_in[2];
    const float* ent_table = (const float*)d_in[3];
    const float* rel_table = (const float*)d_in[4];
    const float* Wr_w      = (const float*)d_in[5];
    const float* Wr_b      = (const float*)d_in[6];
    const float* W1_w      = (const float*)d_in[7];
    const float* W1_b      = (const float*)d_in[8];
    const float* W2_w      = (const float*)d_in[9];
    const float* W2_b      = (const float*)d_in[10];
    float* out = (float*)d_out;

    float* hpN = (float*)d_ws;                       // [B,128]
    float* hmN = hpN + (size_t)BSZ * ED;             // [B,128]

    size_t smemA = (size_t)(128 * PAD + 48 * PAD + 48 * PAD + 32 * PAD +
                            128 + 128 + 32) * sizeof(float) + 64 * sizeof(int);
    kgan_attn<<<BSZ, 128, smemA, stream>>>(idx, adj_ent, adj_rel, ent_table,
                                           rel_table, Wr_w, Wr_b, hpN, hmN);

    size_t smemB = (size_t)(128 * PAD + 64 * PAD + 128) * sizeof(float);
    kgan_out<<<BSZ / 64, 128, smemB, stream>>>(hpN, hmN, W1_w, W1_b, W2_w, W2_b, out);
}